// LSTMNextToken_15101105013344
// MI455X (gfx1250) — compile-verified
//
#include <hip/hip_runtime.h>
#include <math.h>

// Problem constants (match reference)
#define V_    433
#define D_    120
#define B_    4096
#define L_    16
#define H4_   480
#define BL_   (B_ * L_)
#define LN_EPS 1e-5f

typedef __attribute__((ext_vector_type(2))) float v2f;
typedef __attribute__((ext_vector_type(4))) float v4f;
typedef __attribute__((ext_vector_type(8))) float v8f;
typedef __attribute__((ext_vector_type(4))) int   v4i;

// Full-precision fp32 WMMA: D[16x16] += A[16x4] * B[4x16]
__device__ __forceinline__ v8f wmma4(v2f a, v2f b, v8f c) {
  return __builtin_amdgcn_wmma_f32_16x16x4_f32(false, a, false, b,
                                               (short)0, c, false, false);
}

__device__ __forceinline__ float sigf(float x) { return 1.0f / (1.0f + __expf(-x)); }

// ---------------------------------------------------------------------------
// CDNA5 async global->LDS (ASYNCcnt-tracked) with graceful fallback.
// Builtin proto (from probe): (v4i addrspace(1)*, v4i addrspace(3)*, Ii, Ii)
// ---------------------------------------------------------------------------
#if __has_builtin(__builtin_amdgcn_global_load_async_to_lds_b128) && \
    __has_builtin(__builtin_amdgcn_s_wait_asynccnt)
#define USE_ASYNC_LDS 1
#else
#define USE_ASYNC_LDS 0
#endif

__device__ __forceinline__ void async_g2l_b128(const float* g, float* l) {
#if USE_ASYNC_LDS
  __builtin_amdgcn_global_load_async_to_lds_b128(
      (__attribute__((address_space(1))) v4i*)g,
      (__attribute__((address_space(3))) v4i*)l, 0, 0);
#else
  *(v4f*)l = *(const v4f*)g;
#endif
}

__device__ __forceinline__ void async_wait_all() {
#if USE_ASYNC_LDS
  __builtin_amdgcn_s_wait_asynccnt(0);
#endif
}

// ---------------------------------------------------------------------------
// Embedding gather: X0[b,l,:] = embed[x[b,l],:]
// ---------------------------------------------------------------------------
__global__ void embed_gather(const int* __restrict__ x,
                             const float* __restrict__ embed,
                             float* __restrict__ X0) {
  size_t i = (size_t)blockIdx.x * blockDim.x + threadIdx.x;
  const size_t total = (size_t)BL_ * D_;
  const size_t step = (size_t)gridDim.x * blockDim.x;
  for (; i < total; i += step) {
    size_t bl = i / D_;
    int d = (int)(i - bl * D_);
    X0[i] = embed[(size_t)x[bl] * D_ + d];
  }
}

// ---------------------------------------------------------------------------
// Fused LSTM layer. One block owns 16 batch rows and runs all L=16 timesteps.
// gates[16,480] = [x_t | h][16,240] * [Wih | Whh]^T  via fp32 WMMA (K=4).
// x_{t+1} is double-buffered into LDS with async global->LDS loads, overlapped
// with timestep t's WMMA GEMM; per-wave s_wait_asynccnt + barrier publishes it.
// 192 threads = 6 waves; wave w computes n-tiles w*5 .. w*5+4 (30 tiles of 16).
// ---------------------------------------------------------------------------
#define LDP_ 132   // LDS row pitch (floats): 528B, 16B-aligned, pitch%64 = 4

__device__ __forceinline__ void stage_x_async(const float* __restrict__ X,
                                              int b0, int t, float* dst, int tid) {
  // 16 rows x 30 b128 chunks (120 floats/row)
  for (int c = tid; c < 16 * 30; c += 192) {
    const int r = c / 30;
    const int o = c - r * 30;
    async_g2l_b128(X + (size_t)(b0 + r) * (L_ * D_) + t * D_ + o * 4,
                   dst + r * LDP_ + o * 4);
  }
}

__global__ __launch_bounds__(192) void lstm_layer(
    const float* __restrict__ X,    // [B, L, D]
    const float* __restrict__ Wih,  // [H4, D] row-major
    const float* __restrict__ Whh,  // [H4, D] row-major
    const float* __restrict__ bih,  // [H4]
    const float* __restrict__ bhh,  // [H4]
    float* __restrict__ Y)          // [B, L, D]
{
  __shared__ float sX[2][16 * LDP_];  // double-buffered x_t tile
  __shared__ float sH[16 * LDP_];     // h state
  __shared__ float sC[16 * LDP_];     // c state
  __shared__ float sG[16 * 480];      // gates i|f|g|o

  const int tid  = threadIdx.x;
  const int wave = tid >> 5;
  const int lane = tid & 31;
  const int half = lane >> 4;      // K-half select (fp32 WMMA A/B layout)
  const int lrow = lane & 15;      // M (A) / N (B,D) index within tile
  const int b0   = blockIdx.x * 16;

  // Prefetch weights toward L2/L0 (global_prefetch_b8)
  for (int i = tid; i < H4_; i += 192) {
    __builtin_prefetch(Wih + (size_t)i * D_, 0, 0);
    __builtin_prefetch(Whh + (size_t)i * D_, 0, 0);
  }

  // Prologue: kick off async stage of x_0 while zeroing the state
  stage_x_async(X, b0, 0, &sX[0][0], tid);
  for (int i = tid; i < 16 * LDP_; i += 192) { sH[i] = 0.0f; sC[i] = 0.0f; }
  async_wait_all();
  __syncthreads();

  for (int t = 0; t < L_; ++t) {
    const int cur = t & 1;
    // Overlap: issue async stage of x_{t+1} into the other buffer
    if (t + 1 < L_) stage_x_async(X, b0, t + 1, &sX[cur ^ 1][0], tid);

    // Gates GEMM: each wave does 5 n-tiles of 16 gate units
    const float* sXc = &sX[cur][0];
    for (int q = 0; q < 5; ++q) {
      const int n0 = (wave * 5 + q) * 16;
      const int n  = n0 + lrow;
      const float* wih_row = Wih + (size_t)n * D_;
      const float* whh_row = Whh + (size_t)n * D_;
      v8f acc = {};
      // K-part 1: x_t * Wih^T
      for (int k0 = 0; k0 < D_; k0 += 4) {
        const int kk = k0 + 2 * half;
        v2f bf; bf.x = wih_row[kk]; bf.y = wih_row[kk + 1];
        v2f af; af.x = sXc[lrow * LDP_ + kk]; af.y = sXc[lrow * LDP_ + kk + 1];
        acc = wmma4(af, bf, acc);
      }
      // K-part 2: h * Whh^T
      for (int k0 = 0; k0 < D_; k0 += 4) {
        const int kk = k0 + 2 * half;
        v2f bf; bf.x = whh_row[kk]; bf.y = whh_row[kk + 1];
        v2f af; af.x = sH[lrow * LDP_ + kk]; af.y = sH[lrow * LDP_ + kk + 1];
        acc = wmma4(af, bf, acc);
      }
      const float bb = bih[n] + bhh[n];
      #pragma unroll
      for (int r = 0; r < 8; ++r) {
        const int m = r + 8 * half;          // C/D layout: M = r + 8*half
        sG[m * 480 + n] = acc[r] + bb;
      }
    }
    __syncthreads();

    // Elementwise LSTM cell update (PyTorch gate order i,f,g,o)
    for (int i = tid; i < 16 * D_; i += 192) {
      int r = i / D_, j = i - r * D_;
      float gi = sG[r * 480 + j];
      float gf = sG[r * 480 + j + 120];
      float gg = sG[r * 480 + j + 240];
      float go = sG[r * 480 + j + 360];
      float c  = sigf(gf) * sC[r * LDP_ + j] + sigf(gi) * tanhf(gg);
      float h  = sigf(go) * tanhf(c);
      sC[r * LDP_ + j] = c;
      sH[r * LDP_ + j] = h;
      Y[(size_t)(b0 + r) * (L_ * D_) + t * D_ + j] = h;
    }
    // Ensure this wave's async x_{t+1} chunks have landed, then publish
    async_wait_all();
    __syncthreads();
  }
}

// ---------------------------------------------------------------------------
// LayerNorm over D=120, one wave per row (wave32 shuffle reduction)
// ---------------------------------------------------------------------------
__global__ __launch_bounds__(256) void layernorm_k(
    const float* __restrict__ H, const float* __restrict__ gamma,
    const float* __restrict__ beta, float* __restrict__ O)
{
  const int tid  = threadIdx.x;
  const int wave = tid >> 5;
  const int lane = tid & 31;
  const size_t row = (size_t)blockIdx.x * 8 + wave;
  const float* h = H + row * D_;
  const bool has3 = (lane + 96) < D_;   // 120 = 3*32 + 24
  float v0 = h[lane];
  float v1 = h[lane + 32];
  float v2 = h[lane + 64];
  float v3 = has3 ? h[lane + 96] : 0.0f;
  float s = v0 + v1 + v2 + v3;
  for (int off = 16; off; off >>= 1) s += __shfl_xor(s, off, 32);
  const float mu = s * (1.0f / D_);
  float d0 = v0 - mu, d1 = v1 - mu, d2 = v2 - mu, d3 = has3 ? (v3 - mu) : 0.0f;
  float q = d0 * d0 + d1 * d1 + d2 * d2 + d3 * d3;
  for (int off = 16; off; off >>= 1) q += __shfl_xor(q, off, 32);
  const float rstd = rsqrtf(q * (1.0f / D_) + LN_EPS);
  float* o = O + row * D_;
  o[lane]      = d0 * rstd * gamma[lane]      + beta[lane];
  o[lane + 32] = d1 * rstd * gamma[lane + 32] + beta[lane + 32];
  o[lane + 64] = d2 * rstd * gamma[lane + 64] + beta[lane + 64];
  if (has3) o[lane + 96] = d3 * rstd * gamma[lane + 96] + beta[lane + 96];
}

// ---------------------------------------------------------------------------
// Logits GEMM: out[BL,433] = HN[BL,120] * embed^T[120,433] (tied embedding)
// Block: 128-row M tile (async-staged into LDS), full N (28 tiles; pad
// 433->448, masked). 256 threads = 8 waves; wave w owns m-tile w.
// ---------------------------------------------------------------------------
#define LDA_ 124   // 496B pitch: 16B-aligned, pitch%64 = 60 -> conflict-free
__global__ __launch_bounds__(256) void logits_gemm(
    const float* __restrict__ HN, const float* __restrict__ embed,
    float* __restrict__ out)
{
  __shared__ float sA[128 * LDA_];
  const int tid  = threadIdx.x;
  const int wave = tid >> 5;
  const int lane = tid & 31;
  const int half = lane >> 4;
  const int lrow = lane & 15;
  const size_t m0 = (size_t)blockIdx.x * 128;

  // Async-stage the 128x120 A tile (128 rows x 30 b128 chunks)
  for (int c = tid; c < 128 * 30; c += 256) {
    const int r = c / 30;
    const int o = c - r * 30;
    async_g2l_b128(HN + (m0 + r) * D_ + o * 4, &sA[r * LDA_ + o * 4]);
  }
  async_wait_all();
  __syncthreads();

  const int mrow = wave * 16;
  for (int nt = 0; nt < 28; ++nt) {
    const int n = nt * 16 + lrow;
    const bool nval = (n < V_);
    const float* brow = embed + (size_t)n * D_;
    v8f acc = {};
    for (int k0 = 0; k0 < D_; k0 += 4) {
      const int kk = k0 + 2 * half;
      v2f bf;
      bf.x = nval ? brow[kk]     : 0.0f;
      bf.y = nval ? brow[kk + 1] : 0.0f;
      v2f af;
      af.x = sA[(mrow + lrow) * LDA_ + kk];
      af.y = sA[(mrow + lrow) * LDA_ + kk + 1];
      acc = wmma4(af, bf, acc);
    }
    if (nval) {
      #pragma unroll
      for (int r = 0; r < 8; ++r) {
        const int m = mrow + r + 8 * half;
        out[(m0 + m) * V_ + n] = acc[r];
      }
    }
  }
}

// ---------------------------------------------------------------------------
// Host-side launcher
// ---------------------------------------------------------------------------
extern "C" void kernel_launch(void* const* d_in, const int* in_sizes, int n_in,
                              void* d_out, int out_size, void* d_ws, size_t ws_size,
                              hipStream_t stream) {
  const int*   x     = (const int*)  d_in[0];
  const float* embed = (const float*)d_in[1];
  const float* Wih0  = (const float*)d_in[2];
  const float* Whh0  = (const float*)d_in[3];
  const float* bih0  = (const float*)d_in[4];
  const float* bhh0  = (const float*)d_in[5];
  const float* Wih1  = (const float*)d_in[6];
  const float* Whh1  = (const float*)d_in[7];
  const float* bih1  = (const float*)d_in[8];
  const float* bhh1  = (const float*)d_in[9];
  const float* gamma = (const float*)d_in[10];
  const float* beta  = (const float*)d_in[11];
  float* out = (float*)d_out;

  // Workspace: two [BL, D] fp32 buffers (31.5 MB each), ping-ponged.
  float* bufA = (float*)d_ws;                 // X0, then layer-1 output, then LN input
  float* bufB = bufA + (size_t)BL_ * D_;      // layer-0 output, then LN output

  embed_gather<<<512, 256, 0, stream>>>(x, embed, bufA);
  lstm_layer<<<B_ / 16, 192, 0, stream>>>(bufA, Wih0, Whh0, bih0, bhh0, bufB);
  lstm_layer<<<B_ / 16, 192, 0, stream>>>(bufB, Wih1, Whh1, bih1, bhh1, bufA);
  layernorm_k<<<BL_ / 8, 256, 0, stream>>>(bufA, gamma, beta, bufB);
  logits_gemm<<<BL_ / 128, 256, 0, stream>>>(bufB, embed, out);
}